// GCNComplexMoments_21930103013893
// MI455X (gfx1250) — compile-verified
//
#include <hip/hip_runtime.h>
#include <hip/hip_bf16.h>

#define N_NODES 100000
#define N_EDGES 1600000
#define HID 128
#define KDIM 128

typedef __attribute__((ext_vector_type(2))) float v2f;
typedef __attribute__((ext_vector_type(8))) float v8f;

// ---------------------------------------------------------------------------
// Degree / dinv kernels (deg = 1 + indegree, dinv = rsqrt(deg))
// ---------------------------------------------------------------------------
__global__ void deg_init_kernel(float* __restrict__ deg, int n) {
    int i = blockIdx.x * blockDim.x + threadIdx.x;
    if (i < n) deg[i] = 1.0f;
}

__global__ void deg_count_kernel(const int* __restrict__ dst, float* __restrict__ deg, int e) {
    int i = blockIdx.x * blockDim.x + threadIdx.x;
    if (i < e) atomicAdd(&deg[dst[i]], 1.0f);
}

__global__ void deg_finalize_kernel(float* __restrict__ deg, int n) {
    int i = blockIdx.x * blockDim.x + threadIdx.x;
    if (i < n) deg[i] = rsqrtf(deg[i]);
}

// ---------------------------------------------------------------------------
// Rate encoder: r = relu(rates@W1+b1)@W2+b2  (16->8->128), then
// cvec = r @ conv2_W[128:256,:]  (constant term of conv2's linear part).
// Also zeroes the global-mean accumulator for this call.
// ---------------------------------------------------------------------------
__global__ __launch_bounds__(128)
void rates_encode_kernel(const float* __restrict__ rates,
                         const float* __restrict__ W1, const float* __restrict__ b1,
                         const float* __restrict__ W2, const float* __restrict__ b2,
                         const float* __restrict__ conv2_W,   // [256 x 128]
                         float* __restrict__ cvec, float* __restrict__ gsum) {
    __shared__ float e1[8];
    __shared__ float rsh[128];
    const int t = threadIdx.x;
    if (t < 8) {
        float acc = b1[t];
        for (int i = 0; i < 16; ++i) acc += rates[i] * W1[i * 8 + t];
        e1[t] = fmaxf(acc, 0.0f);
    }
    __syncthreads();
    {
        float acc = b2[t];
        #pragma unroll
        for (int j = 0; j < 8; ++j) acc += e1[j] * W2[j * 128 + t];
        rsh[t] = acc;
        gsum[t] = 0.0f;
    }
    __syncthreads();
    {
        float acc = 0.0f;
        for (int k = 0; k < 128; ++k)
            acc += rsh[k] * conv2_W[(size_t)(128 + k) * 128 + t];
        cvec[t] = acc;
    }
}

// ---------------------------------------------------------------------------
// WMMA GEMM: Y[n_rows x 128] = act(X[n_rows x 128]) @ W[128 x 128] (+ addv)
// W is staged in LDS once per block, pre-swizzled so each B fragment
// (W[k][col], W[k+1][col]) is a contiguous 8-byte pair -> one ds_load_b64.
// One wave computes a 32x128 output tile (two A fragments share each B
// fragment -> 2 WMMAs per B load).
//   A frag (16x4): lanes 0-15 M=lane K={0,1}; lanes 16-31 M=lane-16 K={2,3}
//   B frag (4x16): VGPR v, lane-half h -> K = v + 2h, N = lane&15
//   C/D  (16x16):  VGPR r, half h -> M = r + 8h, N = lane&15
// ---------------------------------------------------------------------------
template <bool RELU>
__global__ __launch_bounds__(128)
void gemm_wmma_kernel(const float* __restrict__ X,
                      const float* __restrict__ W,      // 128 x 128 row-major
                      const float* __restrict__ addv,   // nullable [128]
                      float* __restrict__ Y,
                      int n_rows) {
    // Swizzled W: sW[(k>>1)*256 + c*2 + (k&1)]
    __shared__ float sW[KDIM * HID];
    const int t = threadIdx.x;
    #pragma unroll
    for (int j = 0; j < (KDIM * HID / 4) / 128; ++j) {
        const int i = j * 128 + t;                 // float4 index
        const float4 w4 = ((const float4*)W)[i];
        const int k = i >> 5;                      // (i*4)/128
        const int c = (i << 2) & 127;
        float* base = &sW[(k >> 1) * 256 + (k & 1)];
        base[(c + 0) * 2] = w4.x;
        base[(c + 1) * 2] = w4.y;
        base[(c + 2) * 2] = w4.z;
        base[(c + 3) * 2] = w4.w;
    }
    __syncthreads();

    const int wave = blockIdx.x * (blockDim.x >> 5) + (t >> 5);
    const int m0 = wave * 32;
    if (m0 >= n_rows) return;
    const int lane = t & 31;
    const int ml = lane & 15;
    const int h  = lane >> 4;

    v8f acc[16];
    #pragma unroll
    for (int i = 0; i < 16; ++i) acc[i] = {};

    const float* xrow0 = X + (size_t)(m0 + ml) * KDIM;
    const float* xrow1 = X + (size_t)(m0 + 16 + ml) * KDIM;

    #pragma unroll 4
    for (int k0 = 0; k0 < KDIM; k0 += 4) {
        float2 a0v = *(const float2*)(xrow0 + k0 + 2 * h);
        float2 a1v = *(const float2*)(xrow1 + k0 + 2 * h);
        if (RELU) {
            a0v.x = fmaxf(a0v.x, 0.0f); a0v.y = fmaxf(a0v.y, 0.0f);
            a1v.x = fmaxf(a1v.x, 0.0f); a1v.y = fmaxf(a1v.y, 0.0f);
        }
        v2f a0; a0.x = a0v.x; a0.y = a0v.y;
        v2f a1; a1.x = a1v.x; a1.y = a1v.y;
        const float* wb = &sW[((k0 >> 1) + h) * 256 + ml * 2];
        #pragma unroll
        for (int nt = 0; nt < 8; ++nt) {
            const float2 bv = *(const float2*)(wb + nt * 32);
            v2f b; b.x = bv.x; b.y = bv.y;
            acc[nt]     = __builtin_amdgcn_wmma_f32_16x16x4_f32(
                false, a0, false, b, (short)0, acc[nt], false, false);
            acc[8 + nt] = __builtin_amdgcn_wmma_f32_16x16x4_f32(
                false, a1, false, b, (short)0, acc[8 + nt], false, false);
        }
    }

    #pragma unroll
    for (int mt = 0; mt < 2; ++mt) {
        #pragma unroll
        for (int nt = 0; nt < 8; ++nt) {
            const int col = nt * 16 + ml;
            const float add = addv ? addv[col] : 0.0f;
            #pragma unroll
            for (int r = 0; r < 8; ++r)
                Y[(size_t)(m0 + mt * 16 + r + 8 * h) * HID + col] =
                    acc[mt * 8 + nt][r] + add;
        }
    }
}

// ---------------------------------------------------------------------------
// Aggregation: out[n] = bias + H[n]*dinv[n]^2  (self loop + post-agg bias)
// then edge scatter: out[dst] += H[src] * dinv[src]*dinv[dst]
// ---------------------------------------------------------------------------
__global__ void agg_init_kernel(const float* __restrict__ H, const float* __restrict__ dinv,
                                const float* __restrict__ bias, float* __restrict__ out,
                                int n) {
    long long idx = (long long)blockIdx.x * blockDim.x + threadIdx.x;
    int node = (int)(idx >> 7);
    if (node >= n) return;
    int c = (int)(idx & 127);
    float di = dinv[node];
    out[idx] = H[idx] * di * di + bias[c];
}

__global__ void edge_scatter_kernel(const float* __restrict__ H,
                                    const int* __restrict__ src,
                                    const int* __restrict__ dst,
                                    const float* __restrict__ dinv,
                                    float* __restrict__ out, int e) {
    long long idx = (long long)blockIdx.x * blockDim.x + threadIdx.x;
    int ed = (int)(idx >> 5);          // one wave == one edge (32 x float4)
    if (ed >= e) return;
    int q = (int)(idx & 31);
    int s = src[ed], d = dst[ed];
    float w = dinv[s] * dinv[d];
    const float4 hv = ((const float4*)(H + (size_t)s * HID))[q];
    float* o = out + (size_t)d * HID + q * 4;
    atomicAdd(o + 0, hv.x * w);
    atomicAdd(o + 1, hv.y * w);
    atomicAdd(o + 2, hv.z * w);
    atomicAdd(o + 3, hv.w * w);
}

// ---------------------------------------------------------------------------
// Global mean pool (block-partial sums -> 128 atomics per block)
// ---------------------------------------------------------------------------
__global__ __launch_bounds__(128)
void pool_kernel(const float* __restrict__ H, float* __restrict__ gsum, int n) {
    const int c = threadIdx.x;
    const int n0 = blockIdx.x * 250;
    const int n1 = min(n0 + 250, n);
    float acc = 0.0f;
    for (int i = n0; i < n1; ++i) acc += H[(size_t)i * HID + c];
    atomicAdd(&gsum[c], acc);
}

// ---------------------------------------------------------------------------
// Final MLP head: 128 -> 256 (relu) -> 128 (relu) -> 2
// ---------------------------------------------------------------------------
__global__ __launch_bounds__(256)
void mlp_kernel(const float* __restrict__ gsum,
                const float* __restrict__ hid_W, const float* __restrict__ hid_b,
                const float* __restrict__ hid2_W, const float* __restrict__ hid2_b,
                const float* __restrict__ fin_W, const float* __restrict__ fin_b,
                float* __restrict__ out) {
    __shared__ float g[128];
    __shared__ float t1[256];
    __shared__ float t2[128];
    const int t = threadIdx.x;
    if (t < 128) g[t] = gsum[t] * (1.0f / (float)N_NODES);
    __syncthreads();
    {
        float acc = hid_b[t];
        for (int c = 0; c < 128; ++c) acc += g[c] * hid_W[c * 256 + t];
        t1[t] = fmaxf(acc, 0.0f);
    }
    __syncthreads();
    if (t < 128) {
        float acc = hid2_b[t];
        for (int j = 0; j < 256; ++j) acc += t1[j] * hid2_W[j * 128 + t];
        t2[t] = fmaxf(acc, 0.0f);
    }
    __syncthreads();
    if (t < 2) {
        float acc = fin_b[t];
        for (int c = 0; c < 128; ++c) acc += t2[c] * fin_W[c * 2 + t];
        out[t] = acc;
    }
}

// ---------------------------------------------------------------------------
extern "C" void kernel_launch(void* const* d_in, const int* in_sizes, int n_in,
                              void* d_out, int out_size, void* d_ws, size_t ws_size,
                              hipStream_t stream) {
    (void)in_sizes; (void)n_in; (void)out_size; (void)ws_size;
    const float* graph   = (const float*)d_in[0];
    const int*   eidx    = (const int*)d_in[1];
    const float* rates   = (const float*)d_in[2];
    const float* conv1_W = (const float*)d_in[3];
    const float* conv1_b = (const float*)d_in[4];
    const float* enc_W1  = (const float*)d_in[5];
    const float* enc_b1  = (const float*)d_in[6];
    const float* enc_W2  = (const float*)d_in[7];
    const float* enc_b2  = (const float*)d_in[8];
    const float* conv2_W = (const float*)d_in[9];
    const float* conv2_b = (const float*)d_in[10];
    const float* conv3_W = (const float*)d_in[11];
    const float* conv3_b = (const float*)d_in[12];
    const float* hid_W   = (const float*)d_in[13];
    const float* hid_b   = (const float*)d_in[14];
    const float* hid2_W  = (const float*)d_in[15];
    const float* hid2_b  = (const float*)d_in[16];
    const float* fin_W   = (const float*)d_in[17];
    const float* fin_b   = (const float*)d_in[18];

    const int* src = eidx;
    const int* dst = eidx + N_EDGES;

    float* bufA = (float*)d_ws;                       // [N_NODES * HID]
    float* bufB = bufA + (size_t)N_NODES * HID;       // [N_NODES * HID]
    float* dinv = bufB + (size_t)N_NODES * HID;       // [N_NODES]
    float* cvec = dinv + N_NODES;                     // [128]
    float* gsum = cvec + 128;                         // [128]

    const int TPB = 256;
    const dim3 nodeGrid((N_NODES + TPB - 1) / TPB);
    const dim3 edgeGrid((N_EDGES + TPB - 1) / TPB);
    const dim3 nhGrid(((size_t)N_NODES * HID + TPB - 1) / TPB);
    const dim3 scatGrid(((size_t)N_EDGES * 32 + TPB - 1) / TPB);
    const dim3 gemmGrid((N_NODES / 32 + 3) / 4);      // 4 waves/block, 32 rows/wave

    // degree -> dinv (shared by all three convs)
    deg_init_kernel<<<nodeGrid, TPB, 0, stream>>>(dinv, N_NODES);
    deg_count_kernel<<<edgeGrid, TPB, 0, stream>>>(dst, dinv, N_EDGES);
    deg_finalize_kernel<<<nodeGrid, TPB, 0, stream>>>(dinv, N_NODES);

    // rate embedding -> constant conv2 row vector; zero pool accumulator
    rates_encode_kernel<<<1, 128, 0, stream>>>(rates, enc_W1, enc_b1, enc_W2, enc_b2,
                                               conv2_W, cvec, gsum);

    // conv1: lin -> aggregate (bias post-agg); relu deferred to conv2 load
    gemm_wmma_kernel<false><<<gemmGrid, 128, 0, stream>>>(graph, conv1_W, nullptr, bufA, N_NODES);
    agg_init_kernel<<<nhGrid, TPB, 0, stream>>>(bufA, dinv, conv1_b, bufB, N_NODES);
    edge_scatter_kernel<<<scatGrid, TPB, 0, stream>>>(bufA, src, dst, dinv, bufB, N_EDGES);

    // conv2: lin = relu(h1) @ W2[:128] + (r @ W2[128:]) ; aggregate
    gemm_wmma_kernel<true><<<gemmGrid, 128, 0, stream>>>(bufB, conv2_W, cvec, bufA, N_NODES);
    agg_init_kernel<<<nhGrid, TPB, 0, stream>>>(bufA, dinv, conv2_b, bufB, N_NODES);
    edge_scatter_kernel<<<scatGrid, TPB, 0, stream>>>(bufA, src, dst, dinv, bufB, N_EDGES);

    // conv3
    gemm_wmma_kernel<false><<<gemmGrid, 128, 0, stream>>>(bufB, conv3_W, nullptr, bufA, N_NODES);
    agg_init_kernel<<<nhGrid, TPB, 0, stream>>>(bufA, dinv, conv3_b, bufB, N_NODES);
    edge_scatter_kernel<<<scatGrid, TPB, 0, stream>>>(bufA, src, dst, dinv, bufB, N_EDGES);

    // mean pool + MLP head
    pool_kernel<<<dim3((N_NODES + 249) / 250), 128, 0, stream>>>(bufB, gsum, N_NODES);
    mlp_kernel<<<1, 256, 0, stream>>>(gsum, hid_W, hid_b, hid2_W, hid2_b,
                                      fin_W, fin_b, (float*)d_out);
}